// PhysConstrainedLES_14044543058019
// MI455X (gfx1250) — compile-verified
//
#include <hip/hip_runtime.h>
#include <stdint.h>

#define HH 512
#define WW 512
#define OH 510
#define ROWS 8            // output rows per workgroup
#define IN_ROWS (ROWS + 2) // input rows needed (3x3 stencil)
#define NTHREADS 256

// --- CDNA5 async global->LDS copy (ASYNCcnt-tracked, bypasses VGPRs) ---
__device__ __forceinline__ void async_copy16(uint32_t lds_byte_addr, const float* gptr) {
    // vdst = per-lane LDS byte offset, vaddr = per-lane 64-bit global address
    asm volatile("global_load_async_to_lds_b128 %0, %1, off"
                 :
                 : "v"(lds_byte_addr), "v"(gptr)
                 : "memory");
}

__device__ __forceinline__ void wait_async_all() {
    asm volatile("s_wait_asynccnt 0" ::: "memory");
}

__global__ __launch_bounds__(NTHREADS)
void phys_les_kernel(const float* __restrict__ uPred,
                     const float* __restrict__ pPred,
                     float* __restrict__ out)
{
    __shared__ float smem[3 * IN_ROWS * WW];  // 60 KB: u | v | p tiles
    float* u_s = smem;
    float* v_s = smem + IN_ROWS * WW;
    float* p_s = smem + 2 * IN_ROWS * WW;

    const int tid = threadIdx.x;
    const int b   = blockIdx.x >> 6;    // 64 row-blocks per batch
    const int rb  = blockIdx.x & 63;
    const int r0  = rb * ROWS;          // first output row of this block

    const float* ug = uPred + (size_t)(b * 2 + 0) * HH * WW;
    const float* vg = uPred + (size_t)(b * 2 + 1) * HH * WW;
    const float* pg = pPred + (size_t)b * HH * WW;

    const uint32_t lds0 = (uint32_t)(uintptr_t)(void*)smem; // LDS byte offset (addr[31:0])

    // ---- Stage IN_ROWS x 512 f32 of u, v, p into LDS via async b128 copies ----
    // 1280 chunks of 16B per field; 256 threads -> exactly 5 issues/thread/field.
    const int CHUNKS = IN_ROWS * (WW / 4);  // 10 * 128 = 1280
#pragma unroll
    for (int c = tid; c < CHUNKS; c += NTHREADS) {
        const int rr = c >> 7;          // row within tile
        const int cc = c & 127;         // 16B chunk within row
        int gy = r0 + rr; if (gy > HH - 1) gy = HH - 1;  // clamp (unused rows in last block)
        const size_t gofs = (size_t)gy * WW + cc * 4;
        async_copy16(lds0 + (uint32_t)(0 * IN_ROWS * WW * 4) + (uint32_t)c * 16, ug + gofs);
        async_copy16(lds0 + (uint32_t)(1 * IN_ROWS * WW * 4) + (uint32_t)c * 16, vg + gofs);
        async_copy16(lds0 + (uint32_t)(2 * IN_ROWS * WW * 4) + (uint32_t)c * 16, pg + gofs);
    }
    wait_async_all();   // per-wave ASYNCcnt == 0
    __syncthreads();    // all waves' tiles landed

    // Folded stencil constants (match reference's Kx/Ky/Kxx/Kyy scaling exactly)
    const float inv2d  = 50.0f;     // 1/(2*0.01)
    const float invd2  = 10000.0f;  // 1/(0.01*0.01)
    const float dxs    = 0.01f;     // div scale
    const float dxdy   = 1e-4f;     // pstar scale

    float* __restrict__ divOut = out;
    float* __restrict__ psOut  = out + (size_t)32 * OH * WW;

    // ---- div = clip(dx*(dudx + dvdy)) on [510 x 512] (edge-replicated x) ----
#pragma unroll 2
    for (int idx = tid; idx < ROWS * WW; idx += NTHREADS) {
        const int row = idx >> 9;
        const int col = idx & 511;
        const int i = r0 + row;
        if (i >= OH) continue;
        const int cl = (col == 0)   ? 0   : col - 1;
        const int cr = (col == 511) ? 511 : col + 1;
        const float dudx = (u_s[(row + 1) * WW + cr] - u_s[(row + 1) * WW + cl]) * inv2d;
        const float dvdy = (v_s[(row + 2) * WW + col] - v_s[row * WW + col]) * inv2d;
        float d = dxs * (dudx + dvdy);
        d = fminf(fmaxf(d, -1.0f), 1.0f);
        divOut[((size_t)b * OH + i) * WW + col] = d;
    }

    // ---- pstar = clip(dx*dy*(ddp + ux^2 + 2*uy*vx + vy^2)) on [510 x 510] ----
#pragma unroll 2
    for (int idx = tid; idx < ROWS * WW; idx += NTHREADS) {
        const int row = idx >> 9;
        const int col = idx & 511;
        const int i = r0 + row;
        if (i >= OH || col >= 510) continue;
        const float ux = (u_s[(row + 1) * WW + col + 2] - u_s[(row + 1) * WW + col]) * inv2d;
        const float uy = (u_s[(row + 2) * WW + col + 1] - u_s[(row)     * WW + col + 1]) * inv2d;
        const float vx = (v_s[(row + 1) * WW + col + 2] - v_s[(row + 1) * WW + col]) * inv2d;
        const float vy = (v_s[(row + 2) * WW + col + 1] - v_s[(row)     * WW + col + 1]) * inv2d;
        const float pc = p_s[(row + 1) * WW + col + 1];
        const float ddp =
            (p_s[(row + 1) * WW + col]     - 2.0f * pc + p_s[(row + 1) * WW + col + 2]) * invd2 +
            (p_s[(row)     * WW + col + 1] - 2.0f * pc + p_s[(row + 2) * WW + col + 1]) * invd2;
        const float rhs = ux * ux + 2.0f * uy * vx + vy * vy;
        float ps = dxdy * (ddp + rhs);   // rho == 1
        ps = fminf(fmaxf(ps, -1.0f), 1.0f);
        psOut[((size_t)b * OH + i) * 510 + col] = ps;
    }
}

extern "C" void kernel_launch(void* const* d_in, const int* in_sizes, int n_in,
                              void* d_out, int out_size, void* d_ws, size_t ws_size,
                              hipStream_t stream) {
    const float* uPred = (const float*)d_in[0];  // [32,2,512,512] f32
    const float* pPred = (const float*)d_in[1];  // [32,1,512,512] f32
    float* out = (float*)d_out;                  // div(32*510*512) ++ pstar(32*510*510)
    const dim3 grid(32 * 64);                    // 64 row-blocks per batch
    phys_les_kernel<<<grid, NTHREADS, 0, stream>>>(uPred, pPred, out);
}